// RobustListLearner_29497835389129
// MI455X (gfx1250) — compile-verified
//
#include <hip/hip_runtime.h>
#include <stdint.h>

// Problem constants from the reference
#define NN      96        // samples
#define DD      64        // features
#define MS_MAX  4560      // C(96,2)
#define MARGIN  0.1f

typedef __attribute__((ext_vector_type(4))) unsigned int v4u;
typedef __attribute__((ext_vector_type(8))) int          v8i;
typedef __attribute__((ext_vector_type(4))) int          v4i;
typedef __attribute__((ext_vector_type(2))) float        v2f;

// Build a 2D TDM descriptor (data_size = 4B) and issue tensor_load_to_lds.
__device__ __forceinline__ void tdm_load_2d(const void* gptr, unsigned int lds_off,
                                            int dim0, int dim1)
{
    unsigned long long ga = (unsigned long long)(uintptr_t)gptr;

    // D# group 0: count=1 | lds_addr | global_addr[56:0] | type=2 ("image")
    v4u g0;
    g0.x = 1u;                                            // count=1, user mode
    g0.y = lds_off;                                       // lds_addr (bytes)
    g0.z = (unsigned int)(ga & 0xFFFFFFFFull);            // global_addr[31:0]
    g0.w = ((unsigned int)(ga >> 32) & 0x1FFFFFFu)        // global_addr[56:32]
           | (2u << 30);                                  // type=2

    // D# group 1: data_size=4B; tensor dim0 x dim1 == tile dims; stride = dim0
    v8i g1;
    g1[0] = (2 << 16);                 // [17:16] data_size = 2 -> 4 bytes
    g1[1] = (dim0 & 0xFFFF) << 16;     // tensor_dim0[15:0]  -> bits 63:48
    g1[2] = ((dim0 >> 16) & 0xFFFF)    // tensor_dim0[31:16] -> bits 79:64
          | ((dim1 & 0xFFFF) << 16);   // tensor_dim1[15:0]  -> bits 95:80
    g1[3] = (dim0 & 0xFFFF) << 16;     // tile_dim0          -> bits 127:112
    g1[4] = (dim1 & 0xFFFF);           // tile_dim1          -> bits 143:128
    g1[5] = dim0;                      // tensor_dim0_stride[31:0]
    g1[6] = 0;
    g1[7] = 0;

    v4i gz4 = {0, 0, 0, 0};               // groups 2/3 unused (2D tensor)
    v8i gz8 = {0, 0, 0, 0, 0, 0, 0, 0};   // extra group (clang-23 6-arg form)
    __builtin_amdgcn_tensor_load_to_lds(g0, g1, gz4, gz4, gz8, 0);
}

__global__ __launch_bounds__(256)
void robust_list_kernel(const float* __restrict__ labels,
                        const float* __restrict__ features,
                        const int2*  __restrict__ sample_idx,
                        const int2*  __restrict__ feat_idx,
                        float* __restrict__ out_w,      // [Mf*Ms, 2]
                        float* __restrict__ out_idx,    // [Mf*Ms, 2] (as float)
                        int Ms)
{
    __shared__ float smF[NN * DD];    // 24.0 KB: feature matrix, row-major
    __shared__ int2  smS[MS_MAX];     // 35.6 KB: sample index pairs
    __shared__ float smB[NN];         //  384 B : rhs b[i] = 1 - MARGIN*labels[i]

    const int f  = blockIdx.x;
    const int2 pq = feat_idx[f];
    const int p = pq.x, q = pq.y;

    // ---- Tensor Data Mover: two bulk DMAs (global -> LDS), wave 0 issues ----
    if (threadIdx.x == 0) {
        tdm_load_2d(features,
                    (unsigned int)(uintptr_t)(void*)&smF[0], DD, NN);
        tdm_load_2d(sample_idx,
                    (unsigned int)(uintptr_t)(void*)&smS[0], 2 * Ms, 1);
        __builtin_amdgcn_s_wait_tensorcnt(0);             // both DMAs complete
    }

    // rhs staging: labels are +-1  =>  l*(l-MARGIN) = 1 - MARGIN*l
    if (threadIdx.x < NN) smB[threadIdx.x] = 1.0f - MARGIN * labels[threadIdx.x];
    __syncthreads();

    const size_t base = (size_t)f * (size_t)Ms;
    const float fp = (float)p, fq_ = (float)q;

    for (int s = threadIdx.x; s < Ms; s += blockDim.x) {
        const int2 ij = smS[s];                 // ds_load_b64 (LDS-fed loop)
        const int i = ij.x, j = ij.y;

        const float a = smF[i * DD + p];
        const float b = smF[i * DD + q];
        const float c = smF[j * DD + p];
        const float d = smF[j * DD + q];
        const float y0 = smB[i];
        const float y1 = smB[j];

        const float det  = __builtin_fmaf(a, d, -(b * c));
        const float fro2 = a * a + b * b + c * c + d * d;

        float w0, w1;
        if (__builtin_fabsf(det) > 1.1920929e-7f * fro2) {
            // full rank: inverse via adjugate
            const float inv = 1.0f / det;
            w0 = (d * y0 - b * y1) * inv;
            w1 = (a * y1 - c * y0) * inv;
        } else if (fro2 > 0.0f) {
            // rank-1 pseudoinverse: pinv(M) = M^T / ||M||_F^2
            const float inv = 1.0f / fro2;
            w0 = (a * y0 + c * y1) * inv;
            w1 = (b * y0 + d * y1) * inv;
        } else {
            w0 = 0.0f; w1 = 0.0f;
        }

        const size_t o = base + (size_t)s;
        v2f wv;  wv.x = w0;  wv.y = w1;
        v2f iv;  iv.x = fp;  iv.y = fq_;
        // streaming stores: written once, never re-read on device
        __builtin_nontemporal_store(wv, (v2f*)(out_w   + 2 * o));
        __builtin_nontemporal_store(iv, (v2f*)(out_idx + 2 * o));
    }
}

extern "C" void kernel_launch(void* const* d_in, const int* in_sizes, int n_in,
                              void* d_out, int out_size, void* d_ws, size_t ws_size,
                              hipStream_t stream) {
    const float* labels     = (const float*)d_in[0];   // (96,)  f32
    const float* features   = (const float*)d_in[1];   // (96,64) f32
    const int2*  sample_idx = (const int2*)d_in[2];    // (Ms,2) i32
    const int2*  feat_idx   = (const int2*)d_in[3];    // (Mf,2) i32

    const int Ms = in_sizes[2] / 2;   // 4560
    const int Mf = in_sizes[3] / 2;   // 2016

    float* out_w   = (float*)d_out;                         // weights first
    float* out_idx = out_w + (size_t)2 * Mf * (size_t)Ms;   // then col_indices

    robust_list_kernel<<<Mf, 256, 0, stream>>>(labels, features, sample_idx,
                                               feat_idx, out_w, out_idx, Ms);
}